// Sim_vec_Video_8443905704308
// MI455X (gfx1250) — compile-verified
//
#include <hip/hip_runtime.h>
#include <math.h>

// Harness owns main/IO/timing; we implement kernel_launch() + kernels.

typedef __attribute__((ext_vector_type(2))) float v2f;
typedef __attribute__((ext_vector_type(4))) float v4f;
typedef __attribute__((ext_vector_type(8))) float v8f;

#define DIM    1024
#define BATCH  256
#define FRAMES 64
#define EPSV   1e-8f

__device__ __forceinline__ float sigm(float x) {
    return 1.0f / (1.0f + expf(-x));
}

// ---------------- K1: thres[r] = mean_r + sim_w[r] * std_r (ddof=1) ----------------
__global__ void k_thres(const float* __restrict__ emb,
                        const float* __restrict__ simw,
                        float* __restrict__ thres) {
    __shared__ float s_sum[256];
    __shared__ float s_sq[256];
    const int r = blockIdx.x;
    float sum = 0.f, sq = 0.f;
    for (int c = threadIdx.x; c < DIM; c += blockDim.x) {
        float w = sigm(emb[r * DIM + c]);
        sum += w;
        sq  += w * w;
    }
    s_sum[threadIdx.x] = sum;
    s_sq[threadIdx.x]  = sq;
    __syncthreads();
    for (int s = 128; s > 0; s >>= 1) {
        if (threadIdx.x < s) {
            s_sum[threadIdx.x] += s_sum[threadIdx.x + s];
            s_sq[threadIdx.x]  += s_sq[threadIdx.x + s];
        }
        __syncthreads();
    }
    if (threadIdx.x == 0) {
        float mean = s_sum[0] * (1.0f / (float)DIM);
        float var  = (s_sq[0] - (float)DIM * mean * mean) / ((float)DIM - 1.0f);
        var = fmaxf(var, 0.0f);
        thres[r] = mean + simw[r] * sqrtf(var);
    }
}

// ---------------- K2: rownorm[r] = || tanh(exp(et*(w - thres[c]))) * w ||_2 --------
__global__ void k_rownorm(const float* __restrict__ emb,
                          const float* __restrict__ thres,
                          const float* __restrict__ tempw,
                          float* __restrict__ rown) {
    __shared__ float s_sum[256];
    const int r = blockIdx.x;
    const float et = expf(tempw[0]);
    float acc = 0.f;
    for (int c = threadIdx.x; c < DIM; c += blockDim.x) {
        float w = sigm(emb[r * DIM + c]);
        float m = tanhf(expf(et * (w - thres[c]))) * w;
        acc += m * m;
    }
    s_sum[threadIdx.x] = acc;
    __syncthreads();
    for (int s = 128; s > 0; s >>= 1) {
        if (threadIdx.x < s) s_sum[threadIdx.x] += s_sum[threadIdx.x + s];
        __syncthreads();
    }
    if (threadIdx.x == 0) rown[r] = sqrtf(s_sum[0]);
}

// ---------------- K3: diag[c] = sum_r m(r,c) / (rownorm[r] + eps) ------------------
__global__ void k_diag(const float* __restrict__ emb,
                       const float* __restrict__ thres,
                       const float* __restrict__ rown,
                       const float* __restrict__ tempw,
                       float* __restrict__ diag) {
    const int c = blockIdx.x * blockDim.x + threadIdx.x;  // 0..DIM-1
    const float et = expf(tempw[0]);
    const float th = thres[c];
    float acc = 0.f;
    for (int r = 0; r < DIM; ++r) {
        float w = sigm(emb[r * DIM + c]);           // coalesced across threads
        float m = tanhf(expf(et * (w - th))) * w;
        acc += m / (rown[r] + EPSV);
    }
    diag[c] = acc;
}

// ---------------- K4: txtw[i,d] = txt[i,d] * diag[d] -------------------------------
__global__ void k_txtw(const float* __restrict__ txt,
                       const float* __restrict__ diag,
                       float* __restrict__ txtw) {
    const int t = blockIdx.x * blockDim.x + threadIdx.x;
    txtw[t] = txt[t] * diag[t & (DIM - 1)];
}

// ---------------- K5: vbar[j,d] = mean_f vid[j,f,d], float4-vectorized -------------
// Thread t owns 4 consecutive d values -> global_load_b128 / global_store_b128,
// fully coalesced across the wave. This is the 64 MB bandwidth-dominant pass.
__global__ void k_vbar(const float* __restrict__ vid,
                       float* __restrict__ vbar) {
    const int t  = blockIdx.x * blockDim.x + threadIdx.x;  // 0..BATCH*DIM/4-1
    const int j  = t / (DIM / 4);
    const int d4 = (t % (DIM / 4)) * 4;
    const float* p = vid + (size_t)j * FRAMES * DIM + d4;
    v4f s = {0.f, 0.f, 0.f, 0.f};
#pragma unroll
    for (int f = 0; f < FRAMES; ++f) {
        v4f x = *(const v4f*)(p + (size_t)f * DIM);
        s += x;
    }
    s *= (1.0f / (float)FRAMES);
    *(v4f*)(vbar + (size_t)j * DIM + d4) = s;
}

// ---------------- K6: C = txtw @ vbar^T via V_WMMA_F32_16X16X4_F32 -----------------
// One wave per 32x32 output block (2x2 WMMA tiles): 4 independent accumulator
// chains hide the WMMA->WMMA accumulate latency, and each K-step does
// 4 global_load_b64 -> 4 WMMAs (1:1), halving fragment re-read traffic vs 1 tile.
// Fragment layouts (ISA 7.12.2, wave32):
//   A 16x4 f32: lanes 0-15 M=lane, VGPR{0,1}=K{0,1}; lanes 16-31 K{2,3}
//   B 4x16 f32: VGPR v + lane-half h -> K = v + 2h, N = lane&15
//   C/D 16x16 f32: VGPR v, lanes 0-15 M=v, lanes 16-31 M=v+8, N = lane&15
__global__ void k_gemm_wmma(const float* __restrict__ A,   // [BATCH, DIM] txtw
                            const float* __restrict__ Bv,  // [BATCH, DIM] vbar
                            float* __restrict__ C) {       // [BATCH, BATCH]
    const int bi = blockIdx.x >> 3;       // 0..7 (row block of 32)
    const int bj = blockIdx.x & 7;        // 0..7 (col block of 32)
    const int lane = threadIdx.x;         // 0..31 (wave32, EXEC all ones)
    const int lm = lane & 15;
    const int h  = lane >> 4;
    const int kA = h << 1;                // 0 or 2

    const float* a0 = A  + (size_t)(bi * 32 + lm) * DIM + kA;
    const float* a1 = a0 + (size_t)16 * DIM;
    const float* b0 = Bv + (size_t)(bj * 32 + lm) * DIM + kA;
    const float* b1 = b0 + (size_t)16 * DIM;

    v8f c00 = {0.f,0.f,0.f,0.f,0.f,0.f,0.f,0.f};
    v8f c01 = c00, c10 = c00, c11 = c00;

#pragma unroll 2
    for (int k0 = 0; k0 < DIM; k0 += 4) {
        v2f fa0 = *(const v2f*)(a0 + k0);   // global_load_b64
        v2f fa1 = *(const v2f*)(a1 + k0);
        v2f fb0 = *(const v2f*)(b0 + k0);
        v2f fb1 = *(const v2f*)(b1 + k0);
        // 8 args: (neg_a, A, neg_b, B, c_mod, C, reuse_a, reuse_b)
        c00 = __builtin_amdgcn_wmma_f32_16x16x4_f32(false, fa0, false, fb0,
                                                    (short)0, c00, false, false);
        c01 = __builtin_amdgcn_wmma_f32_16x16x4_f32(false, fa0, false, fb1,
                                                    (short)0, c01, false, false);
        c10 = __builtin_amdgcn_wmma_f32_16x16x4_f32(false, fa1, false, fb0,
                                                    (short)0, c10, false, false);
        c11 = __builtin_amdgcn_wmma_f32_16x16x4_f32(false, fa1, false, fb1,
                                                    (short)0, c11, false, false);
    }

    const int r0 = bi * 32 + 8 * h;       // tile-row base for M tiles p=0
    const int cb0 = bj * 32 + lm;         // tile-col base for N tiles q=0
#pragma unroll
    for (int v = 0; v < 8; ++v) {
        C[(size_t)(r0 + v)      * BATCH + cb0     ] = c00[v];
        C[(size_t)(r0 + v)      * BATCH + cb0 + 16] = c01[v];
        C[(size_t)(r0 + 16 + v) * BATCH + cb0     ] = c10[v];
        C[(size_t)(r0 + 16 + v) * BATCH + cb0 + 16] = c11[v];
    }
}

// ---------------- K7: row-wise L2 normalize of sim_all [BATCH, BATCH] --------------
__global__ void k_out_norm(float* __restrict__ out) {
    __shared__ float s_sum[256];
    const int i = blockIdx.x;
    float v = out[i * BATCH + threadIdx.x];
    s_sum[threadIdx.x] = v * v;
    __syncthreads();
    for (int s = 128; s > 0; s >>= 1) {
        if (threadIdx.x < s) s_sum[threadIdx.x] += s_sum[threadIdx.x + s];
        __syncthreads();
    }
    float norm = sqrtf(s_sum[0]) + EPSV;
    out[i * BATCH + threadIdx.x] = v / norm;
}

extern "C" void kernel_launch(void* const* d_in, const int* in_sizes, int n_in,
                              void* d_out, int out_size, void* d_ws, size_t ws_size,
                              hipStream_t stream) {
    (void)in_sizes; (void)n_in; (void)out_size; (void)ws_size;

    const float* txt   = (const float*)d_in[0];   // [256, 1024]
    const float* vid   = (const float*)d_in[1];   // [256, 64, 1024]
    const float* emb   = (const float*)d_in[2];   // [1024, 1024]
    const float* simw  = (const float*)d_in[3];   // [1, 1024]
    const float* tempw = (const float*)d_in[4];   // [1, 1]
    float* out = (float*)d_out;                   // [256, 256]

    float* ws    = (float*)d_ws;
    float* thres = ws;                            // 1024
    float* rown  = ws + 1024;                     // 1024
    float* diag  = ws + 2048;                     // 1024
    float* txtw  = ws + 3072;                     // 256*1024
    float* vbar  = ws + 3072 + BATCH * DIM;       // 256*1024

    k_thres  <<<DIM, 256, 0, stream>>>(emb, simw, thres);
    k_rownorm<<<DIM, 256, 0, stream>>>(emb, thres, tempw, rown);
    k_diag   <<<DIM / 256, 256, 0, stream>>>(emb, thres, rown, tempw, diag);
    k_txtw   <<<(BATCH * DIM) / 256, 256, 0, stream>>>(txt, diag, txtw);
    k_vbar   <<<(BATCH * DIM / 4) / 256, 256, 0, stream>>>(vid, vbar);
    k_gemm_wmma<<<(BATCH / 32) * (BATCH / 32), 32, 0, stream>>>(txtw, vbar, out);
    k_out_norm <<<BATCH, 256, 0, stream>>>(out);
}